// MostSimpleCellEncoder_15891378995346
// MI455X (gfx1250) — compile-verified
//
#include <hip/hip_runtime.h>
#include <hip/hip_bf16.h>
#include <stdint.h>

// Problem constants (from reference setup_inputs)
#define K_NNZ   2048
#define D_DIM   64
#define BIN_LEN 16
#define NWAVES  8
#define PARTS   8                 // K-partitions per batch row -> 512 blocks
#define KP      (K_NNZ / PARTS)   // 256 rows per block
#define PFD     4                 // prefetch distance in per-wave iterations

typedef unsigned int u32x4 __attribute__((ext_vector_type(4)));
typedef int          i32x8 __attribute__((ext_vector_type(8)));
typedef int          i32x4 __attribute__((ext_vector_type(4)));

// Low 32 bits of a flat shared-memory address == LDS byte offset
// (flat LDS aperture: the LDS address is the low 32 bits, see ISA 10.2).
__device__ __forceinline__ unsigned lds_offset_of(const void* p) {
    return (unsigned)(uintptr_t)p;
}

#if __has_builtin(__builtin_amdgcn_tensor_load_to_lds)
// 1-D contiguous DMA of `nelem` 4-byte elements from global into LDS via TDM.
// D# layout per CDNA5 ISA ch.8: group0 = {flags, lds_addr, global_addr, type=2},
// group1 = {data_size=4B, tensor_dim0=nelem, tensor_dim1=1, tile_dim0=nelem,
// tile_dim1=1, dim0_stride=nelem}; groups 2/3 describe degenerate upper dims.
__device__ __forceinline__ void tdm_copy_1d_b32(const void* gsrc, unsigned lds_off, int nelem) {
    uint64_t ga = (uint64_t)(uintptr_t)gsrc;
    u32x4 g0;
    g0.x = 1u;                                   // count=1, user descriptor, no gather
    g0.y = lds_off;                              // lds_addr (bytes)
    g0.z = (unsigned)ga;                         // global_addr[31:0]
    g0.w = (unsigned)((ga >> 32) & 0x01FFFFFFu)  // global_addr[56:32]
         | (2u << 30);                           // type = 2 ("image")
    i32x8 g1;
    g1[0] = (2 << 16);                           // data_size = 2 (4 bytes); no multicast/pad
    g1[1] = (int)(((unsigned)nelem & 0xFFFFu) << 16);   // tensor_dim0[15:0]
    g1[2] = ((nelem >> 16) & 0xFFFF) | (1 << 16);       // tensor_dim0[31:16] | tensor_dim1=1
    g1[3] = (int)(((unsigned)nelem & 0xFFFFu) << 16);   // tensor_dim1[31:16]=0 | tile_dim0
    g1[4] = 1;                                   // tile_dim1 = 1, tile_dim2 = 0
    g1[5] = nelem;                               // tensor_dim0_stride[31:0]
    g1[6] = 0;                                   // stride hi | tensor_dim1_stride lo
    g1[7] = 0;
    i32x4 g2; g2.x = 1; g2.y = 1; g2.z = 0; g2.w = 0;   // tensor_dim2=1, tensor_dim3=1
    i32x4 g3; g3.x = 0; g3.y = (1 << 16); g3.z = 0; g3.w = 0; // tensor_dim4=1, tile_dim4=0
#if defined(__clang_major__) && __clang_major__ >= 23
    i32x8 g4 = {0, 0, 0, 0, 0, 0, 0, 0};
    __builtin_amdgcn_tensor_load_to_lds(g0, g1, g2, g3, g4, 0);
#else
    __builtin_amdgcn_tensor_load_to_lds(g0, g1, g2, g3, 0);
#endif
}
#endif

// Shared device helper: gather + renorm + accumulate `cnt` rows whose indices
// sit in lds_pos[0..cnt), one 256 B row per wave per iteration, lane l owning
// dims {2l, 2l+1}. Returns this thread's float2 partial.
__device__ __forceinline__ float2 gather_renorm_sum(const float* __restrict__ pos_table,
                                                    const int* __restrict__ lds_pos,
                                                    int cnt, int wave, int lane) {
    float2 acc; acc.x = 0.0f; acc.y = 0.0f;
    for (int i = wave; i < cnt; i += NWAVES) {
        const int p = lds_pos[i];
        const float2 v = ((const float2*)(pos_table + (size_t)p * D_DIM))[lane];

        // Software-prefetch a future data-dependent row (global_prefetch_b8).
        if (i + NWAVES * PFD < cnt) {
            const int pn = lds_pos[i + NWAVES * PFD];
            __builtin_prefetch(pos_table + (size_t)pn * D_DIM + 2 * lane, 0, 1);
        }

        float sq = v.x * v.x + v.y * v.y;
#pragma unroll
        for (int off = 16; off > 0; off >>= 1)
            sq += __shfl_xor(sq, off, 32);
        const float n = sqrtf(sq);
        const float s = (n > 1.0f) ? (1.0f / (n + 1e-7f)) : 1.0f;
        acc.x += v.x * s;
        acc.y += v.y * s;
    }
    return acc;
}

// ---------------------------------------------------------------------------
// Stage 1: grid (PARTS, bs) x 256 threads. Each block reduces a 256-row slice
// of one batch row into a per-(b,part) partial vector + partial popcount.
// ---------------------------------------------------------------------------
__global__ __launch_bounds__(256)
void cell_encoder_stage1(const int*   __restrict__ positions,
                         const int*   __restrict__ values,
                         const float* __restrict__ pos_table,
                         float*       __restrict__ partial,   // [bs*PARTS][D_DIM]
                         unsigned*    __restrict__ pc_part)   // [bs*PARTS]
{
    __shared__ int      lds_pos[KP];           // 1 KB slice of feature indices
    __shared__ int      lds_val[KP];           // 1 KB slice of values
    __shared__ float    red[NWAVES][D_DIM];    // 2 KB per-wave partials
    __shared__ unsigned pcnt;

    const int part = blockIdx.x;
    const int b    = blockIdx.y;
    const int tid  = threadIdx.x;
    const int wave = tid >> 5;
    const int lane = tid & 31;
    const size_t base = (size_t)b * K_NNZ + (size_t)part * KP;

    if (tid == 0) pcnt = 0u;

#if __has_builtin(__builtin_amdgcn_tensor_load_to_lds) && \
    __has_builtin(__builtin_amdgcn_s_wait_tensorcnt)
    if (wave == 0) {   // one wave issues both TDM descriptors
        tdm_copy_1d_b32(positions + base, lds_offset_of(lds_pos), KP);
        tdm_copy_1d_b32(values    + base, lds_offset_of(lds_val), KP);
        __builtin_amdgcn_s_wait_tensorcnt(0);
    }
#else
    for (int i = tid; i < KP; i += 256) {
        lds_pos[i] = positions[base + i];
        lds_val[i] = values[base + i];
    }
#endif
    __syncthreads();

    // Partial popcount for this slice (one value per thread).
    unsigned pc = (tid < KP) ? (unsigned)__popc(lds_val[tid] & 0xFFFF) : 0u;
#pragma unroll
    for (int off = 16; off > 0; off >>= 1)
        pc += __shfl_xor(pc, off, 32);
    if (lane == 0) atomicAdd(&pcnt, pc);       // ds_add_u32

    const float2 acc = gather_renorm_sum(pos_table, lds_pos, KP, wave, lane);
    red[wave][2 * lane]     = acc.x;
    red[wave][2 * lane + 1] = acc.y;
    __syncthreads();

    if (tid < D_DIM) {
        float s = 0.0f;
#pragma unroll
        for (int w = 0; w < NWAVES; ++w) s += red[w][tid];
        partial[((size_t)b * PARTS + part) * D_DIM + tid] = s;
    }
    if (tid == 0) pc_part[(size_t)b * PARTS + part] = pcnt;
}

// ---------------------------------------------------------------------------
// Stage 2: grid bs x 64 threads. Deterministic fixed-order reduction over the
// PARTS partials + closed-form value-embedding term (only rows 0/1 of
// val_table are ever indexed, since the "bag" indices are bits).
// ---------------------------------------------------------------------------
__global__ __launch_bounds__(64)
void cell_encoder_stage2(const float*    __restrict__ partial,
                         const unsigned* __restrict__ pc_part,
                         const float*    __restrict__ val_table,
                         float*          __restrict__ out)
{
    const int b = blockIdx.x;
    const int d = threadIdx.x;

    float s = 0.0f;
    unsigned P = 0u;
#pragma unroll
    for (int p = 0; p < PARTS; ++p) {
        s += partial[((size_t)b * PARTS + p) * D_DIM + d];
        P += pc_part[(size_t)b * PARTS + p];
    }

    float n0 = 0.0f, n1 = 0.0f;
    for (int j = 0; j < D_DIM; ++j) {
        const float a = val_table[j];          n0 += a * a;   // row 0
        const float c = val_table[D_DIM + j];  n1 += c * c;   // row 1
    }
    n0 = sqrtf(n0); n1 = sqrtf(n1);
    const float s0 = (n0 > 1.0f) ? (1.0f / (n0 + 1e-7f)) : 1.0f;
    const float s1 = (n1 > 1.0f) ? (1.0f / (n1 + 1e-7f)) : 1.0f;
    const float e0 = val_table[d] * s0;
    const float e1 = val_table[D_DIM + d] * s1;

    const float Pf  = (float)P;
    const float tot = s + Pf * e1 + ((float)(BIN_LEN * K_NNZ) - Pf) * e0;
    out[(size_t)b * D_DIM + d] = tot * (1.0f / (float)K_NNZ);
}

// ---------------------------------------------------------------------------
// Fallback: single-kernel path (one block per batch row), used only if the
// harness workspace is too small for the two-stage partials.
// ---------------------------------------------------------------------------
__global__ __launch_bounds__(256)
void cell_encoder_fused(const int*   __restrict__ positions,
                        const int*   __restrict__ values,
                        const float* __restrict__ pos_table,
                        const float* __restrict__ val_table,
                        float*       __restrict__ out)
{
    __shared__ int      lds_pos[K_NNZ];
    __shared__ int      lds_val[K_NNZ];
    __shared__ float    red[NWAVES][D_DIM];
    __shared__ unsigned pcnt;

    const int b    = blockIdx.x;
    const int tid  = threadIdx.x;
    const int wave = tid >> 5;
    const int lane = tid & 31;

    if (tid == 0) pcnt = 0u;

#if __has_builtin(__builtin_amdgcn_tensor_load_to_lds) && \
    __has_builtin(__builtin_amdgcn_s_wait_tensorcnt)
    if (wave == 0) {
        tdm_copy_1d_b32(positions + (size_t)b * K_NNZ, lds_offset_of(lds_pos), K_NNZ);
        tdm_copy_1d_b32(values    + (size_t)b * K_NNZ, lds_offset_of(lds_val), K_NNZ);
        __builtin_amdgcn_s_wait_tensorcnt(0);
    }
#else
    for (int i = tid; i < K_NNZ; i += 256) {
        lds_pos[i] = positions[(size_t)b * K_NNZ + i];
        lds_val[i] = values[(size_t)b * K_NNZ + i];
    }
#endif
    __syncthreads();

    unsigned pc = 0;
    for (int i = tid; i < K_NNZ; i += 256)
        pc += (unsigned)__popc(lds_val[i] & 0xFFFF);
#pragma unroll
    for (int off = 16; off > 0; off >>= 1)
        pc += __shfl_xor(pc, off, 32);
    if (lane == 0) atomicAdd(&pcnt, pc);

    const float2 acc = gather_renorm_sum(pos_table, lds_pos, K_NNZ, wave, lane);
    red[wave][2 * lane]     = acc.x;
    red[wave][2 * lane + 1] = acc.y;
    __syncthreads();

    if (tid < D_DIM) {
        float s = 0.0f;
#pragma unroll
        for (int w = 0; w < NWAVES; ++w) s += red[w][tid];

        float n0 = 0.0f, n1 = 0.0f;
        for (int j = 0; j < D_DIM; ++j) {
            const float a = val_table[j];          n0 += a * a;
            const float c = val_table[D_DIM + j];  n1 += c * c;
        }
        n0 = sqrtf(n0); n1 = sqrtf(n1);
        const float s0 = (n0 > 1.0f) ? (1.0f / (n0 + 1e-7f)) : 1.0f;
        const float s1 = (n1 > 1.0f) ? (1.0f / (n1 + 1e-7f)) : 1.0f;
        const float e0 = val_table[tid] * s0;
        const float e1 = val_table[D_DIM + tid] * s1;

        const float P   = (float)pcnt;
        const float tot = s + P * e1 + ((float)(BIN_LEN * K_NNZ) - P) * e0;
        out[(size_t)b * D_DIM + tid] = tot * (1.0f / (float)K_NNZ);
    }
}

extern "C" void kernel_launch(void* const* d_in, const int* in_sizes, int n_in,
                              void* d_out, int out_size, void* d_ws, size_t ws_size,
                              hipStream_t stream) {
    const int*   positions = (const int*)d_in[0];
    const int*   values    = (const int*)d_in[1];
    const float* pos_table = (const float*)d_in[2];
    const float* val_table = (const float*)d_in[3];
    float*       out       = (float*)d_out;

    const int bs = in_sizes[0] / K_NNZ;   // 64

    const size_t part_bytes = (size_t)bs * PARTS * D_DIM * sizeof(float);
    const size_t pc_bytes   = (size_t)bs * PARTS * sizeof(unsigned);

    if (ws_size >= part_bytes + pc_bytes) {
        float*    partial = (float*)d_ws;
        unsigned* pc_part = (unsigned*)((char*)d_ws + part_bytes);
        dim3 g1(PARTS, bs);
        cell_encoder_stage1<<<g1, 256, 0, stream>>>(positions, values, pos_table,
                                                    partial, pc_part);
        cell_encoder_stage2<<<bs, D_DIM, 0, stream>>>(partial, pc_part, val_table, out);
    } else {
        cell_encoder_fused<<<bs, 256, 0, stream>>>(positions, values, pos_table,
                                                   val_table, out);
    }
}